// CopyAttentionDistribution_69406671503416
// MI455X (gfx1250) — compile-verified
//
#include <hip/hip_runtime.h>
#include <cstdint>
#include <cstddef>

// Problem constants (match reference: B=32, T=2048, H=1024, V=32000)
#define B_ 32
#define T_ 2048
#define H_ 1024
#define V_ 32000

typedef float v2f __attribute__((ext_vector_type(2)));
typedef float v8f __attribute__((ext_vector_type(8)));

// ---------------------------------------------------------------------------
// Zero the copy_prob region [B,V] and the coverage_loss scalar slot.
// (cov_new region is fully overwritten by k_final, no need to zero.)
// ---------------------------------------------------------------------------
__global__ void k_zero(float* __restrict__ out) {
    size_t i = (size_t)blockIdx.x * blockDim.x + threadIdx.x;
    if (i < (size_t)B_ * V_) out[i] = 0.0f;
    if (i == 0) out[(size_t)B_ * V_ + (size_t)B_ * T_] = 0.0f;
}

// ---------------------------------------------------------------------------
// s[b] = dot(hidden[b,:], Ws) + bs   (tiny: 32 dots of length 1024)
// ---------------------------------------------------------------------------
__global__ __launch_bounds__(256)
void k_hidden(const float* __restrict__ hidden,
              const float* __restrict__ Ws,
              const float* __restrict__ bs,
              float* __restrict__ s_out) {
    __shared__ float red[256];
    const int b = blockIdx.x;
    float acc = 0.0f;
    for (int i = threadIdx.x; i < H_; i += 256)
        acc += hidden[(size_t)b * H_ + i] * Ws[i];
    red[threadIdx.x] = acc;
    __syncthreads();
    for (int off = 128; off > 0; off >>= 1) {
        if (threadIdx.x < off) red[threadIdx.x] += red[threadIdx.x + off];
        __syncthreads();
    }
    if (threadIdx.x == 0) s_out[b] = red[0] + bs[0];
}

// ---------------------------------------------------------------------------
// Heavy kernel: p[b,t] = exp(tanh(s[b] + dot(bert[b,t,:],Wb) + bb + cov[b,t]))
// Dot products via V_WMMA_F32_16X16X4_F32 (full f32 precision, wave32).
//
// Per-wave: a 16-row t-tile. A-matrix 16x4 f32 layout (ISA 7.12.2):
//   lanes 0-15  (M=lane)   : v0=K0, v1=K1  -> load float2 at k+0
//   lanes 16-31 (M=lane-16): v0=K2, v1=K3  -> load float2 at k+2
// B-matrix 4x16: every column replicated with the Wb chunk, rows striped
// across lanes mirroring the C/D half-wave pattern (v0 = k0|k2, v1 = k1|k3).
// Result column N=0 of the 16x16 f32 accumulator: lane 0 holds M=0..7 in
// c[0..7], lane 16 holds M=8..15.
// ---------------------------------------------------------------------------
__global__ __launch_bounds__(256)
void k_logits(const float* __restrict__ bert,
              const float* __restrict__ Wb,
              const float* __restrict__ bb,
              const float* __restrict__ cov,
              const float* __restrict__ s_in,
              float* __restrict__ p_out) {
    __shared__ float lds_wb[H_];
    for (int i = threadIdx.x; i < H_; i += 256) lds_wb[i] = Wb[i];
    __syncthreads();

    const int wave = threadIdx.x >> 5;      // 8 waves / block
    const int lane = threadIdx.x & 31;
    const int m    = lane & 15;             // row within tile
    const int kh   = lane >> 4;             // which K half-pair this lane holds
    const int b    = blockIdx.y;
    const int t0   = blockIdx.x * 128 + wave * 16;

    const float* rowp = bert + ((size_t)b * T_ + (size_t)(t0 + m)) * H_;

    v8f c = {};  // 16x16 f32 accumulator (8 VGPRs)
    #pragma unroll 4
    for (int k = 0; k < H_; k += 4) {
        v2f a = *(const v2f*)(rowp + k + 2 * kh);          // global f32x2
        v2f w = *(const v2f*)(&lds_wb[k + 2 * kh]);        // LDS f32x2 (bcast)
        // 8 args: (neg_a, A, neg_b, B, c_mod, C, reuse_a, reuse_b)
        c = __builtin_amdgcn_wmma_f32_16x16x4_f32(
                false, a, false, w, (short)0, c, false, false);
    }

    // Column N=0 extraction + pointwise math. tanh in [-1,1] -> exp is safe
    // without max-subtraction.
    const float base = s_in[b] + bb[0];
    if (m == 0) {  // lanes 0 (M=0..7) and 16 (M=8..15)
        #pragma unroll
        for (int j = 0; j < 8; ++j) {
            const int t = t0 + kh * 8 + j;
            const float z = tanhf(base + c[j] + cov[(size_t)b * T_ + t]);
            p_out[(size_t)b * T_ + t] = __expf(z);
        }
    }
}

// ---------------------------------------------------------------------------
// sum[b] = sum_t p[b,t]
// ---------------------------------------------------------------------------
__global__ __launch_bounds__(256)
void k_sum(const float* __restrict__ p, float* __restrict__ sum_out) {
    __shared__ float red[256];
    const int b = blockIdx.x;
    float acc = 0.0f;
    for (int t = threadIdx.x; t < T_; t += 256)
        acc += p[(size_t)b * T_ + t];
    red[threadIdx.x] = acc;
    __syncthreads();
    for (int off = 128; off > 0; off >>= 1) {
        if (threadIdx.x < off) red[threadIdx.x] += red[threadIdx.x + off];
        __syncthreads();
    }
    if (threadIdx.x == 0) sum_out[b] = red[0];
}

// ---------------------------------------------------------------------------
// Finalize: attn = p/sum;  cov_new = cov+attn (output);  loss += min(attn,
// cov_new) (block reduce + f32 global atomic);  scatter-add attn into
// copy_prob[b, source[b,t]] via global_atomic_add_f32.
// ---------------------------------------------------------------------------
__global__ __launch_bounds__(256)
void k_final(const float* __restrict__ p,
             const float* __restrict__ sum_in,
             const float* __restrict__ cov,
             const int* __restrict__ source,
             float* __restrict__ out) {
    __shared__ float red[256];
    const size_t idx = (size_t)blockIdx.x * 256 + threadIdx.x;  // < B*T
    const int b = (int)(idx / T_);

    const float attn = p[idx] * (1.0f / sum_in[b]);
    const float cn   = cov[idx] + attn;
    out[(size_t)B_ * V_ + idx] = cn;                      // cov_new output

    atomicAdd(&out[(size_t)b * V_ + (size_t)source[idx]], attn);  // scatter

    red[threadIdx.x] = fminf(attn, cn);
    __syncthreads();
    for (int off = 128; off > 0; off >>= 1) {
        if (threadIdx.x < off) red[threadIdx.x] += red[threadIdx.x + off];
        __syncthreads();
    }
    if (threadIdx.x == 0)
        atomicAdd(&out[(size_t)B_ * V_ + (size_t)B_ * T_], red[0]);
}

// ---------------------------------------------------------------------------
extern "C" void kernel_launch(void* const* d_in, const int* in_sizes, int n_in,
                              void* d_out, int out_size, void* d_ws, size_t ws_size,
                              hipStream_t stream) {
    (void)in_sizes; (void)n_in; (void)out_size; (void)ws_size;

    const float* hidden = (const float*)d_in[0];   // [B,1,H]
    const float* bert   = (const float*)d_in[1];   // [B,T,H]
    const int*   source = (const int*)d_in[2];     // [B,T]
    // d_in[3] = output_size (V), known constant
    const float* cov    = (const float*)d_in[4];   // [B,T,1]
    const float* Ws     = (const float*)d_in[5];   // [H,1]
    const float* bs     = (const float*)d_in[6];   // [1]
    const float* Wb     = (const float*)d_in[7];   // [H,1]
    const float* bb     = (const float*)d_in[8];   // [1]

    float* out = (float*)d_out;   // [B*V] copy_prob | [B*T] cov_new | [1] loss

    // Workspace: p[B*T], s[B], sums[B]  (~257 KB)
    float* ws_p   = (float*)d_ws;
    float* ws_s   = ws_p + (size_t)B_ * T_;
    float* ws_sum = ws_s + B_;

    const size_t nzero = (size_t)B_ * V_;
    k_zero<<<(unsigned)((nzero + 255) / 256), 256, 0, stream>>>(out);

    k_hidden<<<B_, 256, 0, stream>>>(hidden, Ws, bs, ws_s);

    // 8 waves/block, 16 t-rows/wave -> 128 rows/block; grid (T/128, B)
    k_logits<<<dim3(T_ / 128, B_), 256, 0, stream>>>(bert, Wb, bb, cov, ws_s, ws_p);

    k_sum<<<B_, 256, 0, stream>>>(ws_p, ws_sum);

    k_final<<<(B_ * T_) / 256, 256, 0, stream>>>(ws_p, ws_sum, cov, source, out);
}